// InvariantPointAttention_9354438771226
// MI455X (gfx1250) — compile-verified
//
#include <hip/hip_runtime.h>
#include <math.h>

// ---------------- constants ----------------
#define NN    1024
#define HH    12
#define C1    384
#define C2    128
#define COUT  384
#define SQK   16
#define SV    16
#define PQK   4
#define PV    8
#define FINW  2112                 // H*(C2+SV+4*PV)

#define SCALAR_W  0.14433756729740643f   // 1/sqrt(48)
#define POINT_W   0.13608276348795434f   // 1/sqrt(54)
#define W2D_SCALE 0.5773502691896258f    // 1/sqrt(3)

typedef float v2f __attribute__((ext_vector_type(2)));
typedef float v8f __attribute__((ext_vector_type(8)));
typedef int   v4i __attribute__((vector_size(16)));   // matches builtin param

// One f32 WMMA step: D(16x16) += A(16x4) * B(4x16)
__device__ __forceinline__ v8f wmma4(v2f a, v2f b, v8f c) {
  return __builtin_amdgcn_wmma_f32_16x16x4_f32(false, a, false, b, (short)0, c,
                                               false, false);
}

// Async global->LDS 16B copy (CDNA5 GLOBAL_LOAD_ASYNC_TO_LDS_B128, ASYNCcnt),
// with a safe fallback if this toolchain does not declare the builtin.
#if __has_builtin(__builtin_amdgcn_global_load_async_to_lds_b128)
#define HAVE_ASYNC_LDS 1
#else
#define HAVE_ASYNC_LDS 0
#endif

__device__ __forceinline__ void async_copy_b128(const float* gsrc,
                                                float* lds_dst) {
#if HAVE_ASYNC_LDS
  __builtin_amdgcn_global_load_async_to_lds_b128(
      (__attribute__((address_space(1))) v4i*)(v4i*)(void*)(const void*)gsrc,
      (__attribute__((address_space(3))) v4i*)(v4i*)(void*)lds_dst, 0, 0);
#else
  *(float4*)lds_dst = *(const float4*)gsrc;
#endif
}

__device__ __forceinline__ void async_wait0() {
#if HAVE_ASYNC_LDS
#if __has_builtin(__builtin_amdgcn_s_wait_asynccnt)
  __builtin_amdgcn_s_wait_asynccnt(0);
#else
  asm volatile("s_wait_asynccnt 0x0" ::: "memory");
#endif
#endif
}

// =====================================================================
// Kernel 1: generic projection GEMM  out(N x width) = A(N x K) @ W(K x width) + b
// one wave per 16x16 tile
// =====================================================================
__global__ void k_proj_gemm(const float* __restrict__ A,
                            const float* __restrict__ W,
                            const float* __restrict__ bias,
                            float* __restrict__ out, int K, int width) {
  const int lane = threadIdx.x & 31;
  const int m0   = blockIdx.y * 16;
  const int n0   = blockIdx.x * 16;
  const int l15  = lane & 15;
  const int koff = (lane >> 4) * 2;

  v8f acc = {0.f, 0.f, 0.f, 0.f, 0.f, 0.f, 0.f, 0.f};
  const float* Arow = A + (size_t)(m0 + l15) * K;
  for (int k0 = 0; k0 < K; k0 += 4) {
    v2f a; a.x = Arow[k0 + koff]; a.y = Arow[k0 + koff + 1];
    v2f b;
    b.x = W[(size_t)(k0 + koff) * width + n0 + l15];
    b.y = W[(size_t)(k0 + koff + 1) * width + n0 + l15];
    acc = wmma4(a, b, acc);
  }
  const float bv = bias[n0 + l15];
  for (int r = 0; r < 8; ++r) {
    int row = r + 8 * (lane >> 4);
    out[(size_t)(m0 + row) * width + n0 + l15] = acc[r] + bv;
  }
}

// =====================================================================
// Kernel 2: repack projections into head-major layouts + global frames
// block per n
// =====================================================================
__global__ void k_repack(const float* __restrict__ qsf,   // N x 192
                         const float* __restrict__ kvsf,  // N x 384
                         const float* __restrict__ qpf,   // N x 144
                         const float* __restrict__ kvpf,  // N x 432
                         const float* __restrict__ rot,   // N x 9
                         const float* __restrict__ trans, // N x 3
                         float* __restrict__ qs,   // H x N x 16 (scaled)
                         float* __restrict__ ks,   // H x N x 16
                         float* __restrict__ qg,   // H x N x 12
                         float* __restrict__ kg,   // H x N x 12
                         float* __restrict__ vcat, // H x N x 48 (16 vs + 24 vpt + 8 zero)
                         float* __restrict__ q2,   // H x N
                         float* __restrict__ k2) { // H x N
  const int n = blockIdx.x, t = threadIdx.x;
  __shared__ float R[9], T[3];
  __shared__ float GQ[48][3];
  __shared__ float GK[48][3];
  if (t < 9) R[t] = rot[n * 9 + t];
  if (t < 3) T[t] = trans[n * 3 + t];
  __syncthreads();

  for (int idx = t; idx < 192; idx += blockDim.x) {
    int h = idx >> 4, c = idx & 15;
    qs[((size_t)h * NN + n) * 16 + c] = qsf[(size_t)n * 192 + idx] * SCALAR_W;
  }
  for (int idx = t; idx < 384; idx += blockDim.x) {
    int h = idx / 32, c = idx % 32;
    float v = kvsf[(size_t)n * 384 + idx];
    if (c < 16) ks[((size_t)h * NN + n) * 16 + c] = v;
    else        vcat[((size_t)h * NN + n) * 48 + (c - 16)] = v;
  }
  for (int idx = t; idx < HH * 8; idx += blockDim.x) {  // zero pad cols 40..47
    int h = idx / 8, c = 40 + (idx & 7);
    vcat[((size_t)h * NN + n) * 48 + c] = 0.f;
  }
  // q points: flat (n, 144): pts[n, hp, axis] = flat[n, axis*48 + hp]
  for (int hp = t; hp < 48; hp += blockDim.x) {
    float x = qpf[(size_t)n * 144 + hp];
    float y = qpf[(size_t)n * 144 + 48 + hp];
    float z = qpf[(size_t)n * 144 + 96 + hp];
    int h = hp >> 2, p = hp & 3;
    for (int i = 0; i < 3; ++i) {
      float g = R[i * 3 + 0] * x + R[i * 3 + 1] * y + R[i * 3 + 2] * z + T[i];
      GQ[hp][i] = g;
      qg[((size_t)h * NN + n) * 12 + p * 3 + i] = g;
    }
  }
  // kv points: flat (n, 432): pts[n, hpp, axis] = flat[n, axis*144 + hpp]
  for (int hpp = t; hpp < 144; hpp += blockDim.x) {
    float x = kvpf[(size_t)n * 432 + hpp];
    float y = kvpf[(size_t)n * 432 + 144 + hpp];
    float z = kvpf[(size_t)n * 432 + 288 + hpp];
    int h = hpp / 12, pp = hpp % 12;
    for (int i = 0; i < 3; ++i) {
      float g = R[i * 3 + 0] * x + R[i * 3 + 1] * y + R[i * 3 + 2] * z + T[i];
      if (pp < 4) {
        GK[h * 4 + pp][i] = g;
        kg[((size_t)h * NN + n) * 12 + pp * 3 + i] = g;
      } else {
        vcat[((size_t)h * NN + n) * 48 + 16 + (pp - 4) * 3 + i] = g;
      }
    }
  }
  __syncthreads();
  for (int h = t; h < HH; h += blockDim.x) {
    float sq = 0.f, sk = 0.f;
    for (int p = 0; p < 4; ++p)
      for (int i = 0; i < 3; ++i) {
        float a = GQ[h * 4 + p][i]; sq += a * a;
        float b = GK[h * 4 + p][i]; sk += b * b;
      }
    q2[(size_t)h * NN + n] = sq;
    k2[(size_t)h * NN + n] = sk;
  }
}

// =====================================================================
// Kernel 3: logits. block = 16 waves, tile = (16 n) x (16 m), all heads.
// Phase 1: wave w computes a2d tile for row n0+w via WMMA over K=128
//          (first and only pass-1 read of inputs_2d).
// Phase 2: wave h (<12) computes scalar QK^T (K=16) + point cross (K=12)
//          WMMA tiles, combines everything, writes logits.
// =====================================================================
__global__ __launch_bounds__(512) void k_logits(
    const float* __restrict__ in2d, const float* __restrict__ w2d,
    const float* __restrict__ b2d, const float* __restrict__ tpw,
    const float* __restrict__ mask, const float* __restrict__ qs,
    const float* __restrict__ ks, const float* __restrict__ qg,
    const float* __restrict__ kg, const float* __restrict__ q2,
    const float* __restrict__ k2, float* __restrict__ attnS) {
  __shared__ float a2d_s[16][16][17];  // [n_local][m_local][head], padded
  const int wave = threadIdx.x >> 5;
  const int lane = threadIdx.x & 31;
  const int l15  = lane & 15;
  const int koff = (lane >> 4) * 2;
  const int n0 = blockIdx.y * 16;
  const int m0 = blockIdx.x * 16;

  // ---- phase 1: a2d for row n0+wave ----
  {
    const int n = n0 + wave;
    const float* Xrow = in2d + ((size_t)n * NN + m0 + l15) * C2;
    v8f acc = {0.f, 0.f, 0.f, 0.f, 0.f, 0.f, 0.f, 0.f};
    for (int k0 = 0; k0 < C2; k0 += 4) {
      v2f a; a.x = Xrow[k0 + koff]; a.y = Xrow[k0 + koff + 1];
      v2f b;
      b.x = (l15 < HH) ? w2d[(size_t)(k0 + koff) * HH + l15] : 0.f;
      b.y = (l15 < HH) ? w2d[(size_t)(k0 + koff + 1) * HH + l15] : 0.f;
      acc = wmma4(a, b, acc);
    }
    for (int r = 0; r < 8; ++r)
      a2d_s[wave][r + 8 * (lane >> 4)][l15] = acc[r];
  }
  __syncthreads();

  // ---- phase 2: per-head logits ----
  if (wave < HH) {
    const int h = wave;
    const float* qsh = qs + (size_t)h * NN * 16;
    const float* ksh = ks + (size_t)h * NN * 16;
    const float* qgh = qg + (size_t)h * NN * 12;
    const float* kgh = kg + (size_t)h * NN * 12;

    v8f sdot = {0.f, 0.f, 0.f, 0.f, 0.f, 0.f, 0.f, 0.f};
    for (int k0 = 0; k0 < 16; k0 += 4) {
      v2f a; a.x = qsh[(size_t)(n0 + l15) * 16 + k0 + koff];
             a.y = qsh[(size_t)(n0 + l15) * 16 + k0 + koff + 1];
      v2f b; b.x = ksh[(size_t)(m0 + l15) * 16 + k0 + koff];
             b.y = ksh[(size_t)(m0 + l15) * 16 + k0 + koff + 1];
      sdot = wmma4(a, b, sdot);
    }
    v8f cr = {0.f, 0.f, 0.f, 0.f, 0.f, 0.f, 0.f, 0.f};
    for (int k0 = 0; k0 < 12; k0 += 4) {
      v2f a; a.x = qgh[(size_t)(n0 + l15) * 12 + k0 + koff];
             a.y = qgh[(size_t)(n0 + l15) * 12 + k0 + koff + 1];
      v2f b; b.x = kgh[(size_t)(m0 + l15) * 12 + k0 + koff];
             b.y = kgh[(size_t)(m0 + l15) * 12 + k0 + koff + 1];
      cr = wmma4(a, b, cr);
    }

    float tx = tpw[h];
    float pwh = ((tx > 20.f) ? tx : log1pf(expf(tx))) * POINT_W;
    const int m = m0 + l15;
    const float k2v = k2[(size_t)h * NN + m];
    const float mkm = mask[m];
    const float b2h = b2d[h];
    for (int r = 0; r < 8; ++r) {
      int row = r + 8 * (lane >> 4);
      int n = n0 + row;
      float q2v = q2[(size_t)h * NN + n];
      float a2dv = a2d_s[row][l15][h];
      float logit = sdot[r] + pwh * cr[r] - 0.5f * pwh * (q2v + k2v) +
                    W2D_SCALE * (a2dv + b2h);
      float mm = mask[n] * mkm;
      logit -= 100000.f * (1.f - mm);
      attnS[((size_t)h * NN + n) * NN + m] = logit;
    }
  }
}

// =====================================================================
// Kernel 4: row softmax in place, block per (n,h)
// =====================================================================
__global__ void k_softmax(float* __restrict__ attnS) {
  const int n = blockIdx.x, h = blockIdx.y, t = threadIdx.x;
  float* row = attnS + ((size_t)h * NN + n) * NN;
  __shared__ float red[256];
  __shared__ float sb;

  float lm = -3.4e38f;
  for (int m = t; m < NN; m += 256) lm = fmaxf(lm, row[m]);
  red[t] = lm; __syncthreads();
  for (int s = 128; s > 0; s >>= 1) {
    if (t < s) red[t] = fmaxf(red[t], red[t + s]);
    __syncthreads();
  }
  if (t == 0) sb = red[0];
  __syncthreads();
  const float mx = sb;

  float ls = 0.f;
  for (int m = t; m < NN; m += 256) {
    float e = expf(row[m] - mx);
    row[m] = e;
    ls += e;
  }
  __syncthreads();
  red[t] = ls; __syncthreads();
  for (int s = 128; s > 0; s >>= 1) {
    if (t < s) red[t] += red[t + s];
    __syncthreads();
  }
  if (t == 0) sb = 1.f / red[0];
  __syncthreads();
  const float inv = sb;
  for (int m = t; m < NN; m += 256) row[m] *= inv;
}

// =====================================================================
// Kernel 5: attn_2d. block per n (8 waves). Stage attn rows (12x1024) in
// LDS via ASYNC global->LDS b128, then WMMA M=heads(pad16), K=1024, N=128.
// SECOND (last) read of inputs_2d.
// =====================================================================
__global__ __launch_bounds__(256) void k_attn2d(
    const float* __restrict__ attnS, const float* __restrict__ in2d,
    float* __restrict__ finalF) {
  __shared__ float at[12 * 1028];  // padded row stride vs 64 banks
  const int n = blockIdx.x, t = threadIdx.x;
  // 12 rows x 1024 floats = 3072 x 16B segments, 256 lanes -> 12 iters
  for (int idx = t; idx < HH * 256; idx += 256) {
    int h = idx >> 8, seg = idx & 255;
    async_copy_b128(attnS + ((size_t)h * NN + n) * NN + seg * 4,
                    &at[h * 1028 + seg * 4]);
  }
  async_wait0();
  __syncthreads();

  const int wave = t >> 5, lane = t & 31;
  const int l15 = lane & 15;
  const int koff = (lane >> 4) * 2;
  const int c0 = wave * 16;
  const float* X = in2d + (size_t)n * NN * C2;

  v8f acc = {0.f, 0.f, 0.f, 0.f, 0.f, 0.f, 0.f, 0.f};
  for (int k0 = 0; k0 < NN; k0 += 4) {
    if ((k0 & 127) == 0 && k0 + 128 < NN)
      __builtin_prefetch(X + (size_t)(k0 + 128) * C2 + c0 + l15, 0, 3);
    v2f a;
    a.x = (l15 < HH) ? at[l15 * 1028 + k0 + koff] : 0.f;
    a.y = (l15 < HH) ? at[l15 * 1028 + k0 + koff + 1] : 0.f;
    v2f b;
    b.x = X[(size_t)(k0 + koff) * C2 + c0 + l15];
    b.y = X[(size_t)(k0 + koff + 1) * C2 + c0 + l15];
    acc = wmma4(a, b, acc);
  }
  for (int r = 0; r < 8; ++r) {
    int hrow = r + 8 * (lane >> 4);
    if (hrow < HH)
      finalF[(size_t)n * FINW + 576 + hrow * C2 + c0 + l15] = acc[r];
  }
}

// =====================================================================
// Kernel 6: res = attn[h] (NxN) @ vcat[h] (Nx48). One wave computes the
// whole 16x48 output strip (3 accumulators) so attn is read exactly once.
// cols 0..15 -> res_scalar, cols 16..39 -> global point results.
// =====================================================================
__global__ void k_resgemm(const float* __restrict__ attnS,
                          const float* __restrict__ vcat,
                          float* __restrict__ finalF,
                          float* __restrict__ resptg) {
  const int lane = threadIdx.x & 31;
  const int l15 = lane & 15;
  const int koff = (lane >> 4) * 2;
  const int h = blockIdx.y;
  const int n0 = blockIdx.x * 16;
  const float* A = attnS + (size_t)h * NN * NN;
  const float* B = vcat + (size_t)h * NN * 48;

  v8f acc0 = {0.f, 0.f, 0.f, 0.f, 0.f, 0.f, 0.f, 0.f};
  v8f acc1 = acc0, acc2 = acc0;
  const float* Arow = A + (size_t)(n0 + l15) * NN;
  for (int k0 = 0; k0 < NN; k0 += 4) {
    v2f a; a.x = Arow[k0 + koff]; a.y = Arow[k0 + koff + 1];
    const float* Br0 = B + (size_t)(k0 + koff) * 48;
    const float* Br1 = B + (size_t)(k0 + koff + 1) * 48;
    v2f b0; b0.x = Br0[l15];      b0.y = Br1[l15];
    v2f b1; b1.x = Br0[16 + l15]; b1.y = Br1[16 + l15];
    v2f b2; b2.x = Br0[32 + l15]; b2.y = Br1[32 + l15];
    acc0 = wmma4(a, b0, acc0);
    acc1 = wmma4(a, b1, acc1);
    acc2 = wmma4(a, b2, acc2);
  }
  for (int r = 0; r < 8; ++r) {
    int n = n0 + r + 8 * (lane >> 4);
    finalF[(size_t)n * FINW + h * SV + l15] = acc0[r];          // cols 0..15
    resptg[((size_t)n * HH + h) * 24 + l15] = acc1[r];          // cols 16..31
    if (l15 < 8)
      resptg[((size_t)n * HH + h) * 24 + 16 + l15] = acc2[r];   // cols 32..39
  }
}

// =====================================================================
// Kernel 7: local-frame transform + dist. block per n, thread per (h,p).
// =====================================================================
__global__ void k_localxf(const float* __restrict__ resptg,
                          const float* __restrict__ rot,
                          const float* __restrict__ trans,
                          float* __restrict__ finalF) {
  const int n = blockIdx.x, t = threadIdx.x;
  __shared__ float R[9], T[3];
  if (t < 9) R[t] = rot[n * 9 + t];
  if (t < 3) T[t] = trans[n * 3 + t];
  __syncthreads();
  if (t >= HH * PV) return;
  const int h = t >> 3, p = t & 7;
  float d0 = resptg[((size_t)n * HH + h) * 24 + p * 3 + 0] - T[0];
  float d1 = resptg[((size_t)n * HH + h) * 24 + p * 3 + 1] - T[1];
  float d2 = resptg[((size_t)n * HH + h) * 24 + p * 3 + 2] - T[2];
  // einsum 'nji,npj->npi': l_i = sum_j R[j,i] * d_j
  float l0 = R[0] * d0 + R[3] * d1 + R[6] * d2;
  float l1 = R[1] * d0 + R[4] * d1 + R[7] * d2;
  float l2 = R[2] * d0 + R[5] * d1 + R[8] * d2;
  float dist = sqrtf(1e-8f + l0 * l0 + l1 * l1 + l2 * l2);
  const size_t base = (size_t)n * FINW + 192;
  finalF[base + 0 * 96 + t] = l0;
  finalF[base + 1 * 96 + t] = l1;
  finalF[base + 2 * 96 + t] = l2;
  finalF[base + 3 * 96 + t] = dist;
}

// =====================================================================
// Kernel 8: output GEMM  out(N x 384) = finalF(N x 2112) @ w_out + b_out.
// Each wave produces 4 column tiles (64 cols) so A is read once per 4 tiles.
// =====================================================================
__global__ void k_outgemm(const float* __restrict__ F,
                          const float* __restrict__ W,
                          const float* __restrict__ bias,
                          float* __restrict__ out) {
  const int lane = threadIdx.x & 31;
  const int l15 = lane & 15;
  const int koff = (lane >> 4) * 2;
  const int n0 = blockIdx.y * 16;
  const int c0 = blockIdx.x * 64;

  v8f acc[4];
  for (int j = 0; j < 4; ++j)
    acc[j] = (v8f){0.f, 0.f, 0.f, 0.f, 0.f, 0.f, 0.f, 0.f};
  const float* Frow = F + (size_t)(n0 + l15) * FINW;
  for (int k0 = 0; k0 < FINW; k0 += 4) {
    v2f a; a.x = Frow[k0 + koff]; a.y = Frow[k0 + koff + 1];
    const float* Wr0 = W + (size_t)(k0 + koff) * COUT + c0 + l15;
    const float* Wr1 = W + (size_t)(k0 + koff + 1) * COUT + c0 + l15;
    for (int j = 0; j < 4; ++j) {
      v2f b; b.x = Wr0[j * 16]; b.y = Wr1[j * 16];
      acc[j] = wmma4(a, b, acc[j]);
    }
  }
  for (int j = 0; j < 4; ++j) {
    const float bv = bias[c0 + j * 16 + l15];
    for (int r = 0; r < 8; ++r) {
      int n = n0 + r + 8 * (lane >> 4);
      out[(size_t)n * COUT + c0 + j * 16 + l15] = acc[j][r] + bv;
    }
  }
}

// =====================================================================
extern "C" void kernel_launch(void* const* d_in, const int* in_sizes, int n_in,
                              void* d_out, int out_size, void* d_ws,
                              size_t ws_size, hipStream_t stream) {
  const float* inputs_1d = (const float*)d_in[0];
  const float* inputs_2d = (const float*)d_in[1];
  const float* mask      = (const float*)d_in[2];
  const float* rot       = (const float*)d_in[3];
  const float* trans     = (const float*)d_in[4];
  const float* w_q_s     = (const float*)d_in[5];
  const float* b_q_s     = (const float*)d_in[6];
  const float* w_kv_s    = (const float*)d_in[7];
  const float* b_kv_s    = (const float*)d_in[8];
  const float* w_q_p     = (const float*)d_in[9];
  const float* b_q_p     = (const float*)d_in[10];
  const float* w_kv_p    = (const float*)d_in[11];
  const float* b_kv_p    = (const float*)d_in[12];
  const float* w_2d      = (const float*)d_in[13];
  const float* b_2d      = (const float*)d_in[14];
  const float* tpw       = (const float*)d_in[15];
  const float* w_out     = (const float*)d_in[16];
  const float* b_out     = (const float*)d_in[17];
  float* out = (float*)d_out;

  float* ws = (float*)d_ws;
  size_t o = 0;
  float* attnS = ws + o; o += (size_t)HH * NN * NN;   // 48 MB
  float* qsf   = ws + o; o += (size_t)NN * 192;
  float* kvsf  = ws + o; o += (size_t)NN * 384;
  float* qpf   = ws + o; o += (size_t)NN * 144;
  float* kvpf  = ws + o; o += (size_t)NN * 432;
  float* qs    = ws + o; o += (size_t)HH * NN * 16;
  float* ks    = ws + o; o += (size_t)HH * NN * 16;
  float* qg    = ws + o; o += (size_t)HH * NN * 12;
  float* kg    = ws + o; o += (size_t)HH * NN * 12;
  float* vcat  = ws + o; o += (size_t)HH * NN * 48;
  float* q2    = ws + o; o += (size_t)HH * NN;
  float* k2    = ws + o; o += (size_t)HH * NN;
  float* resptg= ws + o; o += (size_t)NN * HH * 24;
  float* finalF= ws + o; o += (size_t)NN * FINW;

  // 1) projections
  k_proj_gemm<<<dim3(192 / 16, NN / 16), 32, 0, stream>>>(inputs_1d, w_q_s,  b_q_s,  qsf,  C1, 192);
  k_proj_gemm<<<dim3(384 / 16, NN / 16), 32, 0, stream>>>(inputs_1d, w_kv_s, b_kv_s, kvsf, C1, 384);
  k_proj_gemm<<<dim3(144 / 16, NN / 16), 32, 0, stream>>>(inputs_1d, w_q_p,  b_q_p,  qpf,  C1, 144);
  k_proj_gemm<<<dim3(432 / 16, NN / 16), 32, 0, stream>>>(inputs_1d, w_kv_p, b_kv_p, kvpf, C1, 432);
  // 2) repack + frames
  k_repack<<<NN, 192, 0, stream>>>(qsf, kvsf, qpf, kvpf, rot, trans, qs, ks, qg,
                                   kg, vcat, q2, k2);
  // 3) logits (first read of inputs_2d, fused a2d)
  k_logits<<<dim3(NN / 16, NN / 16), 512, 0, stream>>>(
      inputs_2d, w_2d, b_2d, tpw, mask, qs, ks, qg, kg, q2, k2, attnS);
  // 4) softmax
  k_softmax<<<dim3(NN, HH), 256, 0, stream>>>(attnS);
  // 5) attn @ inputs_2d (second read of inputs_2d)
  k_attn2d<<<NN, 256, 0, stream>>>(attnS, inputs_2d, finalF);
  // 6) attn @ [v_scalar | v_point]
  k_resgemm<<<dim3(NN / 16, HH), 32, 0, stream>>>(attnS, vcat, finalF, resptg);
  // 7) local frame + dist
  k_localxf<<<NN, 96, 0, stream>>>(resptg, rot, trans, finalF);
  // 8) output projection
  k_outgemm<<<dim3(COUT / 64, NN / 16), 32, 0, stream>>>(finalF, w_out, b_out, out);
}